// Solver_22548578304618
// MI455X (gfx1250) — compile-verified
//
#include <hip/hip_runtime.h>
#include <stdint.h>

typedef __attribute__((ext_vector_type(2))) float v2f;
typedef __attribute__((ext_vector_type(8))) float v8f;

enum { MODE_CHEB = 0, MODE_JACOBI = 1, MODE_RESID = 2 };

#define TSX 64
#define TSY 16
#define MROWS 4
#define HTW (TSX + 2)
#define HTH (TSY + 2)

__device__ __forceinline__ int iclampi(int v, int lo, int hi) {
  return v < lo ? lo : (v > hi ? hi : v);
}

// Faithful float32 reproduction of _polar_coeffs(H, W, i)
__device__ __forceinline__ void polar_coefs(int i, int H, int W,
                                            float& cu, float& cd, float& ct, float& cc) {
  float fH = (float)H, fW = (float)W, fi = (float)i;
  float dr  = 1.0f / fH;
  float dth = 6.28318530717958647692f / fW;
  float r   = fi * dr;
  float rp  = r + 0.5f * dr;
  float rm  = r - 0.5f * dr;
  cu = rp / (r * dr * dr);
  cd = rm / (r * dr * dr);
  ct = 1.0f / (r * r * dth * dth);
  cc = -cu - cd - 2.0f * ct;
}

// conv_pad(u)[y][x] for any (y, x): edge-pad of conv2d_polar core + passthrough last row.
__device__ __forceinline__ float conv_pad_at(const float* __restrict__ u, int y, int x,
                                             int H, int W) {
  int xc = iclampi(x, 1, W - 2);
  int r  = iclampi(y - 1, 0, H - 3);
  if (r == H - 3) return u[(H - 2) * W + xc];  // passthrough row
  int i = r + 1;
  float cu, cd, ct, cc;
  polar_coefs(i, H, W, cu, cd, ct, cc);
  const float* rm1 = u + (i - 1) * W;
  const float* rc  = u + i * W;
  const float* rp1 = u + (i + 1) * W;
  return cu * rm1[xc] + cd * rp1[xc] + ct * (rc[xc - 1] + rc[xc + 1]) + cc * rc[xc];
}

// conv_pad at (y, x) and (y, x+1), x even, sharing row decode + coefficients.
__device__ __forceinline__ void conv_pad_pair(const float* __restrict__ u, int y, int x,
                                              int H, int W, float& cp0, float& cp1) {
  int xc0 = iclampi(x, 1, W - 2);
  int xc1 = iclampi(x + 1, 1, W - 2);
  int r = iclampi(y - 1, 0, H - 3);
  if (r == H - 3) {
    const float* rr = u + (H - 2) * W;
    cp0 = rr[xc0];
    cp1 = rr[xc1];
    return;
  }
  int i = r + 1;
  float cu, cd, ct, cc;
  polar_coefs(i, H, W, cu, cd, ct, cc);
  const float* rm1 = u + (i - 1) * W;
  const float* rc  = u + i * W;
  const float* rp1 = u + (i + 1) * W;
  cp0 = cu * rm1[xc0] + cd * rp1[xc0] + ct * (rc[xc0 - 1] + rc[xc0 + 1]) + cc * rc[xc0];
  cp1 = cu * rm1[xc1] + cd * rp1[xc1] + ct * (rc[xc1 - 1] + rc[xc1 + 1]) + cc * rc[xc1];
}

// Fused stencil sweep: CHEB  : out = u + (f - conv_pad(u)) * omega[ok]/(-4 W^2)
//                      JACOBI: out = bc(u + w*(f - conv_pad(u))/c_c(y))
//                      RESID : out = f - conv_pad(u)
// Interior tiles stage a (TSY+2)x(TSX+2) halo of u into LDS via async global->LDS DMA.
template <int MODE>
__global__ void __launch_bounds__(256)
k_stencil(const float* __restrict__ u, const float* __restrict__ f,
          float* __restrict__ out, int H, int W, float wgt,
          const float* __restrict__ omega, int ok) {
  __shared__ float tile[HTH * HTW];
  const int tx  = threadIdx.x;           // 0..63
  const int ty  = threadIdx.y;           // 0..3
  const int tid = ty * TSX + tx;
  const int gx0 = blockIdx.x * TSX;
  const int gy0 = blockIdx.y * TSY;

  float scale = 0.0f;
  if (MODE == MODE_CHEB) scale = omega[ok] * (-0.25f) / ((float)W * (float)W);

  // Fast path valid when every output point is a regular interior stencil:
  // rows 1..H-3 (i == y) and cols 1..W-2 (xc == x); halo then stays in-bounds.
  const bool fast = (gy0 >= 1) && (gy0 + TSY - 1 <= H - 3) &&
                    (gx0 >= 1) && (gx0 + TSX - 1 <= W - 2);

  if (fast) {
    for (int idx = tid; idx < HTH * HTW; idx += 256) {
      int ly = idx / HTW;
      int lx = idx - ly * HTW;
      const float* gp = u + (gy0 - 1 + ly) * W + (gx0 - 1 + lx);
#if __has_builtin(__builtin_amdgcn_global_load_async_to_lds_b32)
      typedef __attribute__((address_space(1))) int as1int;
      typedef __attribute__((address_space(3))) int as3int;
      __builtin_amdgcn_global_load_async_to_lds_b32(
          (as1int*)(uintptr_t)gp,
          (as3int*)(uint32_t)(uintptr_t)(&tile[idx]), 0, 0);
#else
      tile[idx] = *gp;
#endif
    }
    // warm L2/L0 with the f tile while the async DMA runs
    __builtin_prefetch(&f[(gy0 + ty * MROWS) * W + gx0 + tx], 0, 1);
#if __has_builtin(__builtin_amdgcn_s_wait_asynccnt)
    __builtin_amdgcn_s_wait_asynccnt(0);
#else
    asm volatile("s_wait_asynccnt 0" ::: "memory");
#endif
    __syncthreads();

    const int x = gx0 + tx;
#pragma unroll
    for (int m = 0; m < MROWS; ++m) {
      const int y = gy0 + ty * MROWS + m;
      float cu, cd, ct, cc;
      polar_coefs(y, H, W, cu, cd, ct, cc);
      const int lr = (y - gy0) + 1;
      const float um = tile[(lr - 1) * HTW + tx + 1];
      const float uc = tile[lr * HTW + tx + 1];
      const float ul = tile[lr * HTW + tx];
      const float ur = tile[lr * HTW + tx + 2];
      const float up = tile[(lr + 1) * HTW + tx + 1];
      const float cp = cu * um + cd * up + ct * (ul + ur) + cc * uc;
      const float fv = f[y * W + x];
      float v;
      if (MODE == MODE_CHEB)        v = uc + (fv - cp) * scale;
      else if (MODE == MODE_JACOBI) v = uc + wgt * (fv - cp) / cc;
      else                          v = fv - cp;
      out[y * W + x] = v;
    }
  } else {
    const int x = gx0 + tx;
#pragma unroll
    for (int m = 0; m < MROWS; ++m) {
      const int y = gy0 + ty * MROWS + m;
      if (y >= H || x >= W) continue;
      float v;
      if (MODE == MODE_CHEB) {
        float cp = conv_pad_at(u, y, x, H, W);
        v = u[y * W + x] + (f[y * W + x] - cp) * scale;
      } else if (MODE == MODE_JACOBI) {
        if (y == 0 || y == H - 1) {
          v = 0.0f;                          // _apply_bc rows
        } else {
          int xe = (x == W - 1) ? 0 : x;     // _apply_bc: col W-1 := col 0
          float cp = conv_pad_at(u, y, xe, H, W);
          float cu, cd, ct, cc;
          polar_coefs(y, H, W, cu, cd, ct, cc);
          v = u[y * W + xe] + wgt * (f[y * W + xe] - cp) / cc;
        }
      } else {
        float cp = conv_pad_at(u, y, x, H, W);
        v = f[y * W + x] - cp;
      }
      out[y * W + x] = v;
    }
  }
}

// jacobi with u == 0 (conv_pad(0) == 0): out = bc(w * f / c_c(y))
__global__ void k_jacobi0(const float* __restrict__ f, float* __restrict__ out,
                          int H, int W, int lgW, float wgt) {
  int o = blockIdx.x * blockDim.x + threadIdx.x;
  if (o >= H * W) return;
  int y = o >> lgW, x = o & (W - 1);
  float v = 0.0f;
  if (y > 0 && y < H - 1) {
    int xe = (x == W - 1) ? 0 : x;
    float cu, cd, ct, cc;
    polar_coefs(y, H, W, cu, cd, ct, cc);
    v = wgt * f[y * W + xe] / cc;
  }
  out[o] = v;
}

// first chebyshev iteration with u == 0: out = f * omega[ok]/(-4 W^2)
__global__ void k_cheb0(const float* __restrict__ f, float* __restrict__ out,
                        int N, int W, const float* __restrict__ omega, int ok) {
  int o = blockIdx.x * blockDim.x + threadIdx.x;
  if (o >= N) return;
  float scale = omega[ok] * (-0.25f) / ((float)W * (float)W);
  out[o] = f[o] * scale;
}

// bilinear restriction (H,W) -> (H/2,W/2) with _apply_bc; lgW = log2(W)
__global__ void k_restrict(const float* __restrict__ r, float* __restrict__ out,
                           int H, int W, int lgW) {
  int oh = H >> 1, ow = W >> 1;
  int o = blockIdx.x * blockDim.x + threadIdx.x;
  if (o >= oh * ow) return;
  int yo = o >> (lgW - 1), xo = o & (ow - 1);
  float v = 0.0f;
  if (yo > 0 && yo < oh - 1) {
    int xe = (xo == ow - 1) ? 0 : xo;
    float py = (float)yo * ((float)(H - 1) / (float)(oh - 1));
    float px = (float)xe * ((float)(W - 1) / (float)(ow - 1));
    int i0 = (int)py; if (i0 > H - 2) i0 = H - 2;
    int j0 = (int)px; if (j0 > W - 2) j0 = W - 2;
    float wy = py - (float)i0, wx = px - (float)j0;
    const float* r0 = r + i0 * W + j0;
    const float* r1 = r0 + W;
    v = (r0[0] * (1.0f - wx) + r0[1] * wx) * (1.0f - wy) +
        (r1[0] * (1.0f - wx) + r1[1] * wx) * wy;
  }
  out[o] = v;
}

// out(H,W) += _apply_bc(bilinear-upsample(uc(H/2,W/2))); lgW = log2(W)
__global__ void k_prolong_add(const float* __restrict__ uc, float* __restrict__ out,
                              int H, int W, int lgW) {
  int h = H >> 1, w = W >> 1;
  int o = blockIdx.x * blockDim.x + threadIdx.x;
  if (o >= H * W) return;
  int y = o >> lgW, x = o & (W - 1);
  if (y == 0 || y == H - 1) return;          // bc rows contribute 0
  int xe = (x == W - 1) ? 0 : x;             // bc col copy
  float py = (float)y * ((float)(h - 1) / (float)(H - 1));
  float px = (float)xe * ((float)(w - 1) / (float)(W - 1));
  int i0 = (int)py; if (i0 > h - 2) i0 = h - 2;
  int j0 = (int)px; if (j0 > w - 2) j0 = w - 2;
  float wy = py - (float)i0, wx = px - (float)j0;
  const float* r0 = uc + i0 * w + j0;
  const float* r1 = r0 + w;
  float v = (r0[0] * (1.0f - wx) + r0[1] * wx) * (1.0f - wy) +
            (r1[0] * (1.0f - wx) + r1[1] * wx) * wy;
  out[o] += v;
}

// Deterministic two-sum reduction: sum|f - conv_pad(u)| and sum|f|.
// Per-wave accumulation runs on the fp32 matrix pipe: D += ones(16x4) x B(4x16)
// sums 64 elements per v_wmma_f32_16x16x4_f32 (every row of D = column sums).
__global__ void __launch_bounds__(256)
k_resnorm(const float* __restrict__ u, const float* __restrict__ f,
          float* __restrict__ part, int H, int W, int lgW, int nwaves, int iters) {
  __shared__ float sredR[8], sredF[8];
  const int tid   = threadIdx.x;
  const int wv    = tid >> 5;
  const int lane  = tid & 31;
  const int gwave = blockIdx.x * 8 + wv;

  v8f accR = {0.f, 0.f, 0.f, 0.f, 0.f, 0.f, 0.f, 0.f};
  v8f accF = {0.f, 0.f, 0.f, 0.f, 0.f, 0.f, 0.f, 0.f};
  v2f ones; ones.x = 1.0f; ones.y = 1.0f;

  for (int t = 0; t < iters; ++t) {
    int e = (t * nwaves + gwave) * 64 + lane * 2;   // e even -> pair in same row
    int y = e >> lgW, x = e & (W - 1);
    float2 fv = *(const float2*)(f + e);            // 8B-aligned (e even)
    float cp0, cp1;
    conv_pad_pair(u, y, x, H, W, cp0, cp1);
    v2f br; br.x = fabsf(fv.x - cp0); br.y = fabsf(fv.y - cp1);
    v2f bf; bf.x = fabsf(fv.x);       bf.y = fabsf(fv.y);
#if __has_builtin(__builtin_amdgcn_wmma_f32_16x16x4_f32)
    accR = __builtin_amdgcn_wmma_f32_16x16x4_f32(false, ones, false, br,
                                                 (short)0, accR, false, false);
    accF = __builtin_amdgcn_wmma_f32_16x16x4_f32(false, ones, false, bf,
                                                 (short)0, accF, false, false);
#else
    accR[0] += br.x + br.y;
    accF[0] += bf.x + bf.y;
#endif
  }

#if __has_builtin(__builtin_amdgcn_wmma_f32_16x16x4_f32)
  // lane's accX[0] = column sum; each column sum appears twice across the wave
  float sr = accR[0] * 0.5f;
  float sf = accF[0] * 0.5f;
#else
  float sr = accR[0];
  float sf = accF[0];
#endif
  for (int m = 16; m >= 1; m >>= 1) {
    sr += __shfl_xor(sr, m, 32);
    sf += __shfl_xor(sf, m, 32);
  }
  if (lane == 0) { sredR[wv] = sr; sredF[wv] = sf; }
  __syncthreads();
  if (tid == 0) {
    float tr = 0.0f, tf = 0.0f;
    for (int i = 0; i < 8; ++i) { tr += sredR[i]; tf += sredF[i]; }
    part[blockIdx.x * 2]     = tr;
    part[blockIdx.x * 2 + 1] = tf;
  }
}

__global__ void k_finalize(const float* __restrict__ part, int nb,
                           float* __restrict__ res) {
  if (blockIdx.x == 0 && threadIdx.x == 0) {
    float sr = 0.0f, sf = 0.0f;
    for (int i = 0; i < nb; ++i) { sr += part[2 * i]; sf += part[2 * i + 1]; }
    res[0] = sr / sf;
  }
}

extern "C" void kernel_launch(void* const* d_in, const int* in_sizes, int n_in,
                              void* d_out, int out_size, void* d_ws, size_t ws_size,
                              hipStream_t stream) {
  const float* f     = (const float*)d_in[0];   // (1,1,2048,2048)
  const float* omega = (const float*)d_in[1];   // (5,10)
  float* uout = (float*)d_out;                  // 2048*2048 u, then 1 residual

  const int H0 = 2048, W0 = 2048, N0 = H0 * W0, LG0 = 11;
  const int H1 = 1024, W1 = 1024, N1 = H1 * W1, LG1 = 10;
  const int H2 = 512,  W2 = 512,  N2 = H2 * W2, LG2 = 9;

  float* ws   = (float*)d_ws;
  float* R0   = ws;            // N0
  float* U0   = R0 + N0;       // N0
  float* U1a  = U0 + N0;       // N1
  float* U1b  = U1a + N1;      // N1
  float* F1   = U1b + N1;      // N1
  float* U2   = F1 + N1;       // N2
  float* F2   = U2 + N2;       // N2
  float* part = F2 + N2;       // 1024

  const int TPB = 256;
  dim3 sb(TSX, 4);
  dim3 sg0(W0 / TSX, H0 / TSY), sg1(W1 / TSX, H1 / TSY);

  // ---- idx 0 (op=-1, lvl 0): jacobi(U0=0) -> residual -> restrict -> F1
  k_jacobi0<<<(N0 + TPB - 1) / TPB, TPB, 0, stream>>>(f, U0, H0, W0, LG0, 0.8f);
  k_stencil<MODE_RESID><<<sg0, sb, 0, stream>>>(U0, f, R0, H0, W0, 0.0f, nullptr, 0);
  k_restrict<<<(N1 + TPB - 1) / TPB, TPB, 0, stream>>>(R0, F1, H0, W0, LG0);

  // ---- idx 1 (op=-1, lvl 1): chebyshev x10 (U1=0) -> residual -> restrict -> F2
  k_cheb0<<<(N1 + TPB - 1) / TPB, TPB, 0, stream>>>(F1, U1b, N1, W1, omega, 10 * 1 + 0);
  {
    float* src = U1b; float* dst = U1a;
    for (int k = 1; k < 10; ++k) {               // 9 iters: result lands in U1a
      k_stencil<MODE_CHEB><<<sg1, sb, 0, stream>>>(src, F1, dst, H1, W1, 0.0f,
                                                   omega, 10 * 1 + k);
      float* t = src; src = dst; dst = t;
    }
  }
  k_stencil<MODE_RESID><<<sg1, sb, 0, stream>>>(U1a, F1, U1b, H1, W1, 0.0f, nullptr, 0);
  k_restrict<<<(N2 + TPB - 1) / TPB, TPB, 0, stream>>>(U1b, F2, H1, W1, LG1);

  // ---- idx 2 (op=0, lvl 2): jacobi(U2=0)
  k_jacobi0<<<(N2 + TPB - 1) / TPB, TPB, 0, stream>>>(F2, U2, H2, W2, LG2, 0.8f);

  // ---- idx 3 (op=+1): U1 += prolongate(U2); chebyshev x10
  k_prolong_add<<<(N1 + TPB - 1) / TPB, TPB, 0, stream>>>(U2, U1a, H1, W1, LG1);
  {
    float* src = U1a; float* dst = U1b;
    for (int k = 0; k < 10; ++k) {               // 10 iters: result back in U1a
      k_stencil<MODE_CHEB><<<sg1, sb, 0, stream>>>(src, F1, dst, H1, W1, 0.0f,
                                                   omega, 10 * 3 + k);
      float* t = src; src = dst; dst = t;
    }
  }

  // ---- idx 4 (op=+1): U0 += prolongate(U1); jacobi -> final u in d_out
  k_prolong_add<<<(N0 + TPB - 1) / TPB, TPB, 0, stream>>>(U1a, U0, H0, W0, LG0);
  k_stencil<MODE_JACOBI><<<sg0, sb, 0, stream>>>(U0, f, uout, H0, W0, 0.8f, nullptr, 0);

  // ---- residual norm: sum|f - conv_pad(u)| / sum|f| -> d_out[N0]
  const int NB = 512;                            // 512 blocks * 8 waves * 64 elem * 16 it = N0
  k_resnorm<<<NB, 256, 0, stream>>>(uout, f, part, H0, W0, LG0, NB * 8,
                                    N0 / (NB * 8 * 64));
  k_finalize<<<1, 32, 0, stream>>>(part, NB, uout + N0);
}